// HypergraphPointerNet_24292335026573
// MI455X (gfx1250) — compile-verified
//
#include <hip/hip_runtime.h>
#include <hip/hip_bf16.h>

// Problem constants (from reference)
#define NN   20000      // nodes
#define NE   2048       // hyperedges
#define NNZ  320000     // incidences
#define BB   4          // batch
#define HH   64         // hidden = state dim
#define MROWS (BB * NN) // 80000 flattened GEMM rows

typedef __attribute__((ext_vector_type(2))) float v2f;
typedef __attribute__((ext_vector_type(8))) float v8f;

// ---------------------------------------------------------------------------
// C[M x 64] = A[M x 64] @ W[64 x 64], all fp32 row-major, via V_WMMA_F32_16X16X4_F32.
// One wave per 16-row strip; 4 column tiles of 16; 16 K-steps of 16x16x4.
// A-frag (ISA 7.12.2, 32-bit A 16x4): lanes 0-15 hold K={0,1}, lanes 16-31 K={2,3}.
// B/C/D: row striped across lanes (VGPR r: lanes 0-15 -> M=r, lanes 16-31 -> M=8+r).
// ---------------------------------------------------------------------------
__global__ __launch_bounds__(256)
void gemm64_wmma(float* __restrict__ C, const float* __restrict__ A,
                 const float* __restrict__ W, int M) {
  const int lane = threadIdx.x & 31;
  const int wave = threadIdx.x >> 5;
  const int rbase = (blockIdx.x * 8 + wave) * 16;
  if (rbase >= M) return;
  const int half = lane >> 4;   // 0 -> K pair {0,1}; 1 -> K pair {2,3}
  const int l15  = lane & 15;

  // Hoist all A fragments for the 64-deep K loop (16 x v2f = 32 VGPRs).
  const float* arow = A + (size_t)(rbase + l15) * HH;
  v2f afrag[16];
#pragma unroll
  for (int ks = 0; ks < 16; ++ks) {
    const int k = 4 * ks + 2 * half;          // even -> 8B aligned
    afrag[ks] = *(const v2f*)(arow + k);
  }

#pragma unroll
  for (int nt = 0; nt < 4; ++nt) {
    v8f acc = {};
    const int n = nt * 16 + l15;
#pragma unroll
    for (int ks = 0; ks < 16; ++ks) {
      const int k = 4 * ks + 2 * half;
      v2f bfrag;
      bfrag.x = W[(size_t)k * HH + n];
      bfrag.y = W[(size_t)(k + 1) * HH + n];
      acc = __builtin_amdgcn_wmma_f32_16x16x4_f32(
          /*neg_a=*/false, afrag[ks], /*neg_b=*/false, bfrag,
          /*c_mod=*/(short)0, acc, /*reuse_a=*/false, /*reuse_b=*/false);
    }
    // D layout: VGPR r -> row rbase + r + 8*half, col nt*16 + l15
#pragma unroll
    for (int r = 0; r < 8; ++r) {
      C[(size_t)(rbase + r + 8 * half) * HH + n] = acc[r];
    }
  }
}

// ---------------------------------------------------------------------------
// Degree accumulation: b_cnt[e] += 1 ; d[n] += w[e]   (float atomics, L2-resident)
// ---------------------------------------------------------------------------
__global__ __launch_bounds__(256)
void degrees_kernel(float* __restrict__ bcnt, float* __restrict__ d,
                    const int* __restrict__ nidx, const int* __restrict__ eidx,
                    const int* __restrict__ wm, int nnz) {
  const int i = blockIdx.x * blockDim.x + threadIdx.x;
  if (i >= nnz) return;
  const int e = eidx[i];
  atomicAdd(&bcnt[e], 1.0f);
  atomicAdd(&d[nidx[i]], (float)wm[e]);
}

__global__ __launch_bounds__(256)
void invert_deg(float* __restrict__ binv, float* __restrict__ dinv) {
  const int i = blockIdx.x * blockDim.x + threadIdx.x;
  if (i < NE) { const float v = binv[i]; binv[i] = v > 0.f ? 1.f / v : 0.f; }
  if (i < NN) { const float v = dinv[i]; dinv[i] = v > 0.f ? 1.f / v : 0.f; }
}

// ---------------------------------------------------------------------------
// node -> edge gather: ef[e, b, h] += xl[b, n, h].  One wave per incidence;
// each lane owns a contiguous 8-float slice of the [4,64] tile.
// ---------------------------------------------------------------------------
__global__ __launch_bounds__(256)
void scatter_to_edge(float* __restrict__ ef, const float* __restrict__ xl,
                     const int* __restrict__ nidx, const int* __restrict__ eidx,
                     int nnz) {
  const int gw = (blockIdx.x * blockDim.x + threadIdx.x) >> 5;
  if (gw >= nnz) return;
  const int lane = threadIdx.x & 31;
  const int n = nidx[gw], e = eidx[gw];
  const int b = lane >> 3;
  const int h = (lane & 7) << 3;
  const float* src = xl + ((size_t)b * NN + n) * HH + h;
  float* dst = ef + ((size_t)e * BB + b) * HH + h;
  const float4 v0 = *(const float4*)src;
  const float4 v1 = *(const float4*)(src + 4);
  atomicAdd(dst + 0, v0.x); atomicAdd(dst + 1, v0.y);
  atomicAdd(dst + 2, v0.z); atomicAdd(dst + 3, v0.w);
  atomicAdd(dst + 4, v1.x); atomicAdd(dst + 5, v1.y);
  atomicAdd(dst + 6, v1.z); atomicAdd(dst + 7, v1.w);
}

// ---------------------------------------------------------------------------
// edge -> node scatter with B^{-1} and edge weight folded in:
// out[b, n, h] += ef[e, b, h] * (b_inv[e] * w[e])
// ---------------------------------------------------------------------------
__global__ __launch_bounds__(256)
void scatter_to_node(float* __restrict__ out, const float* __restrict__ ef,
                     const int* __restrict__ nidx, const int* __restrict__ eidx,
                     const float* __restrict__ binv, const int* __restrict__ wm,
                     int nnz) {
  const int gw = (blockIdx.x * blockDim.x + threadIdx.x) >> 5;
  if (gw >= nnz) return;
  const int lane = threadIdx.x & 31;
  const int n = nidx[gw], e = eidx[gw];
  const float s = binv[e] * (float)wm[e];
  const int b = lane >> 3;
  const int h = (lane & 7) << 3;
  const float* src = ef + ((size_t)e * BB + b) * HH + h;
  float* dst = out + ((size_t)b * NN + n) * HH + h;
  const float4 v0 = *(const float4*)src;
  const float4 v1 = *(const float4*)(src + 4);
  atomicAdd(dst + 0, v0.x * s); atomicAdd(dst + 1, v0.y * s);
  atomicAdd(dst + 2, v0.z * s); atomicAdd(dst + 3, v0.w * s);
  atomicAdd(dst + 4, v1.x * s); atomicAdd(dst + 5, v1.y * s);
  atomicAdd(dst + 6, v1.z * s); atomicAdd(dst + 7, v1.w * s);
}

// h = relu(acc * d_inv[n] + bias[h]) in place
__global__ __launch_bounds__(256)
void conv_epilogue(float* __restrict__ buf, const float* __restrict__ dinv,
                   const float* __restrict__ bias, int total) {
  const int i = blockIdx.x * blockDim.x + threadIdx.x;
  if (i >= total) return;
  const int h = i & (HH - 1);
  const int node = (i >> 6) % NN;
  const float v = buf[i] * dinv[node] + bias[h];
  buf[i] = v > 0.f ? v : 0.f;
}

// graph_emb[b, h] = sum_n h2[b, n, h]  (hierarchical: per-block partial + atomic)
__global__ __launch_bounds__(64)
void graph_sum(float* __restrict__ gsum, const float* __restrict__ h2) {
  const int numChunks = (NN + 511) / 512;   // 40
  const int b = blockIdx.x / numChunks;
  const int chunk = blockIdx.x % numChunks;
  const int hc = threadIdx.x;               // 64 threads, one per feature
  const int n0 = chunk * 512;
  const int n1 = (n0 + 512 < NN) ? n0 + 512 : NN;
  float acc = 0.f;
  const float* base = h2 + (size_t)b * NN * HH + hc;
  for (int n = n0; n < n1; ++n) acc += base[(size_t)n * HH];
  atomicAdd(&gsum[b * HH + hc], acc);
}

// gW2[b, h'] = sum_h gsum[b, h] * W2[h, h']
__global__ __launch_bounds__(256)
void gw2_kernel(float* __restrict__ gw2, const float* __restrict__ gsum,
                const float* __restrict__ W2) {
  const int t = threadIdx.x;                // 256 = B*H
  const int b = t >> 6, hp = t & 63;
  float acc = 0.f;
  for (int h = 0; h < HH; ++h) acc += gsum[b * HH + h] * W2[h * HH + hp];
  gw2[t] = acc;
}

// logits[b, n] = sum_k tanh(t[b, n, k] + gW2[b, k]) * w3[k]
__global__ __launch_bounds__(256)
void pointer_final(float* __restrict__ out, const float* __restrict__ t,
                   const float* __restrict__ gw2, const float* __restrict__ w3,
                   int total) {
  const int i = blockIdx.x * blockDim.x + threadIdx.x;
  if (i >= total) return;
  const int b = i / NN;
  const float* trow = t + (size_t)i * HH;
  const float* g = gw2 + b * HH;
  float acc = 0.f;
#pragma unroll 8
  for (int k = 0; k < HH; ++k) acc += tanhf(trow[k] + g[k]) * w3[k];
  out[i] = acc;
}

// ---------------------------------------------------------------------------
extern "C" void kernel_launch(void* const* d_in, const int* in_sizes, int n_in,
                              void* d_out, int out_size, void* d_ws, size_t ws_size,
                              hipStream_t stream) {
  (void)in_sizes; (void)n_in; (void)out_size; (void)ws_size;
  const float* state    = (const float*)d_in[0];
  const int*   hidx     = (const int*)d_in[1];
  const int*   node_idx = hidx;            // hyperedge_index[0]
  const int*   edge_idx = hidx + NNZ;      // hyperedge_index[1]
  const int*   wm       = (const int*)d_in[2];
  const float* w0       = (const float*)d_in[3];
  const float* b0       = (const float*)d_in[4];
  const float* w1       = (const float*)d_in[5];
  const float* b1       = (const float*)d_in[6];
  const float* W1       = (const float*)d_in[7];
  const float* W2       = (const float*)d_in[8];
  const float* w3       = (const float*)d_in[9];
  float* out = (float*)d_out;

  // Workspace layout (floats): 3 activation buffers + edge features + degrees
  float* ws   = (float*)d_ws;
  float* XL   = ws;                         // 5,120,000  (Theta output / reused as t)
  float* H1   = ws + 5120000;               // 5,120,000
  float* H2   = ws + 10240000;              // 5,120,000
  float* EF   = ws + 15360000;              //   524,288  (edge features [E,B,H])
  float* BINV = ws + 15884288;              //     2,048  (b_cnt -> b_inv)
  float* DINV = ws + 15886336;              //    20,000  (d -> d_inv)
  float* GSUM = ws + 15906336;              //       256
  float* GW2  = ws + 15906592;              //       256

  const int TPB = 256;
  const int nnzBlocks   = (NNZ + TPB - 1) / TPB;          // 1250
  const int waveBlocks  = (NNZ * 32 + TPB - 1) / TPB;     // 40000 (wave per nnz)
  const int gemmBlocks  = MROWS / (8 * 16);               // 625
  const int elemBlocks  = (MROWS * HH) / TPB;             // 20000
  const int rowBlocks   = (MROWS + TPB - 1) / TPB;        // 313

  // --- degrees (shared across both layers) ---
  hipMemsetAsync(BINV, 0, NE * sizeof(float), stream);
  hipMemsetAsync(DINV, 0, NN * sizeof(float), stream);
  degrees_kernel<<<nnzBlocks, TPB, 0, stream>>>(BINV, DINV, node_idx, edge_idx, wm, NNZ);
  invert_deg<<<(NN + TPB - 1) / TPB, TPB, 0, stream>>>(BINV, DINV);

  // --- layer 1 ---
  gemm64_wmma<<<gemmBlocks, TPB, 0, stream>>>(XL, state, w0, MROWS);
  hipMemsetAsync(EF, 0, (size_t)NE * BB * HH * sizeof(float), stream);
  scatter_to_edge<<<waveBlocks, TPB, 0, stream>>>(EF, XL, node_idx, edge_idx, NNZ);
  hipMemsetAsync(H1, 0, (size_t)MROWS * HH * sizeof(float), stream);
  scatter_to_node<<<waveBlocks, TPB, 0, stream>>>(H1, EF, node_idx, edge_idx, BINV, wm, NNZ);
  conv_epilogue<<<elemBlocks, TPB, 0, stream>>>(H1, DINV, b0, MROWS * HH);

  // --- layer 2 ---
  gemm64_wmma<<<gemmBlocks, TPB, 0, stream>>>(XL, H1, w1, MROWS);
  hipMemsetAsync(EF, 0, (size_t)NE * BB * HH * sizeof(float), stream);
  scatter_to_edge<<<waveBlocks, TPB, 0, stream>>>(EF, XL, node_idx, edge_idx, NNZ);
  hipMemsetAsync(H2, 0, (size_t)MROWS * HH * sizeof(float), stream);
  scatter_to_node<<<waveBlocks, TPB, 0, stream>>>(H2, EF, node_idx, edge_idx, BINV, wm, NNZ);
  conv_epilogue<<<elemBlocks, TPB, 0, stream>>>(H2, DINV, b1, MROWS * HH);

  // --- pointer scoring ---
  hipMemsetAsync(GSUM, 0, BB * HH * sizeof(float), stream);
  graph_sum<<<BB * ((NN + 511) / 512), 64, 0, stream>>>(GSUM, H2);
  gw2_kernel<<<1, TPB, 0, stream>>>(GW2, GSUM, W2);
  gemm64_wmma<<<gemmBlocks, TPB, 0, stream>>>(XL, H2, W1, MROWS);  // t = h @ W1
  pointer_final<<<rowBlocks, TPB, 0, stream>>>(out, XL, GW2, w3, MROWS);
}